// QSP_Circ_Fit_28845000360455
// MI455X (gfx1250) — compile-verified
//
#include <hip/hip_runtime.h>

#define BLOCK 256
#define SPT   8          // independent samples per thread (ILP for FMA chains)
#define MAXPHI 64

__global__ __launch_bounds__(BLOCK) void qsp_chain_kernel(
    const float* __restrict__ phi,
    const float* __restrict__ wre,   // [N,2,2] row-major, a = wre[4n+0]
    const float* __restrict__ wim,   // [N,2,2] row-major, s = wim[4n+1]
    float* __restrict__ out,
    int n, int nphi)
{
    __shared__ float  lphi[MAXPHI];
    __shared__ float2 trig[MAXPHI];   // (cos(phi/2), sin(phi/2))

    const int tid = threadIdx.x;

    // ---- Stage phi into LDS via CDNA5 async global->LDS path ----
#if __has_builtin(__builtin_amdgcn_global_load_async_to_lds_b32)
    if (tid < nphi) {
        __builtin_amdgcn_global_load_async_to_lds_b32(
            (__attribute__((address_space(1))) int*)(phi + tid),
            (__attribute__((address_space(3))) int*)(lphi + tid),
            0, 0);
    }
  #if __has_builtin(__builtin_amdgcn_s_wait_asynccnt)
    __builtin_amdgcn_s_wait_asynccnt(0);
  #endif
#else
    if (tid < nphi) lphi[tid] = phi[tid];
#endif
    __syncthreads();

    // ---- Precompute wave-uniform RZ trig table once per block ----
    if (tid < nphi) {
        float h = 0.5f * lphi[tid];
        trig[tid] = make_float2(cosf(h), sinf(h));
    }
    __syncthreads();

    // ---- Per-sample state: v = column 0 of M, init (1/sqrt2, 1/sqrt2) ----
    const float RS2 = 0.70710678118654752440f;
    long base = (long)blockIdx.x * (BLOCK * SPT) + tid;

    float a[SPT], s[SPT], x0[SPT], y0[SPT], x1[SPT], y1[SPT];
#pragma unroll
    for (int j = 0; j < SPT; ++j) {
        long idx = base + (long)j * BLOCK;
        bool ok = idx < n;
        a[j] = ok ? wre[idx * 4 + 0] : 0.0f;   // W diagonal (real)
        s[j] = ok ? wim[idx * 4 + 1] : 0.0f;   // W off-diagonal (imag), exact bits
        x0[j] = RS2; y0[j] = 0.0f;
        x1[j] = RS2; y1[j] = 0.0f;
    }

    const int deg = nphi - 1;
    for (int k = 0; k < deg; ++k) {
        float2 cs = trig[k];            // ds_load_b64, uniform -> broadcast
        float c = cs.x, sn = cs.y;

        // RZ(phi_k): v0 *= (c - i sn), v1 *= (c + i sn).
        // Wave-uniform skip when it is the identity (common case here).
        if (sn != 0.0f || c != 1.0f) {
#pragma unroll
            for (int j = 0; j < SPT; ++j) {
                float nx0 = c * x0[j] + sn * y0[j];
                float ny0 = c * y0[j] - sn * x0[j];
                float nx1 = c * x1[j] - sn * y1[j];
                float ny1 = c * y1[j] + sn * x1[j];
                x0[j] = nx0; y0[j] = ny0; x1[j] = nx1; y1[j] = ny1;
            }
        }

        // W = [[a, i s],[i s, a]]:  v0' = a v0 + i s v1 ;  v1' = i s v0 + a v1
#pragma unroll
        for (int j = 0; j < SPT; ++j) {
            float nx0 = a[j] * x0[j] - s[j] * y1[j];
            float ny0 = a[j] * y0[j] + s[j] * x1[j];
            float nx1 = a[j] * x1[j] - s[j] * y0[j];
            float ny1 = a[j] * y1[j] + s[j] * x0[j];
            x0[j] = nx0; y0[j] = ny0; x1[j] = nx1; y1[j] = ny1;
        }
    }

    // Final RZ(phi_deg), then H; output = Re(U[0,0]) = (Re v0 + Re v1)/sqrt2
    {
        float2 cs = trig[deg];
        float c = cs.x, sn = cs.y;
#pragma unroll
        for (int j = 0; j < SPT; ++j) {
            long idx = base + (long)j * BLOCK;
            if (idx < n) {
                float re0 = c * x0[j] + sn * y0[j];   // Re(v0 * (c - i sn))
                float re1 = c * x1[j] - sn * y1[j];   // Re(v1 * (c + i sn))
                out[idx] = RS2 * (re0 + re1);
            }
        }
    }
}

extern "C" void kernel_launch(void* const* d_in, const int* in_sizes, int n_in,
                              void* d_out, int out_size, void* d_ws, size_t ws_size,
                              hipStream_t stream) {
    const float* phi = (const float*)d_in[0];
    const float* wre = (const float*)d_in[1];
    const float* wim = (const float*)d_in[2];
    float* out = (float*)d_out;

    int nphi = in_sizes[0];           // 64
    int n    = out_size;              // 1048576 samples

    int per_block = BLOCK * SPT;
    int grid = (n + per_block - 1) / per_block;
    qsp_chain_kernel<<<grid, BLOCK, 0, stream>>>(phi, wre, wim, out, n, nphi);
}